// KPlaneDensityField_71382356460121
// MI455X (gfx1250) — compile-verified
//
#include <hip/hip_runtime.h>
#include <hip/hip_bf16.h>

// ---------------------------------------------------------------------------
// K-Planes density field, MI455X (gfx1250, wave32).
//   sigma = exp( (prod_i bilerp(plane_i, p)) @ (w1 @ w2) )
// Gather-latency-bound: planes (~11MB) are L2-resident (192MB L2); HBM traffic
// ~95MB -> ~4us floor @ 23.3TB/s. Optimization: repack planes [2,H,W] ->
// [H,W,2] so each bilerp corner is one b64 load (24 loads/pt vs 48).
// The only true matmul (c = w1@w2, 2x64 * 64x1) is done with
// V_WMMA_F32_16X16X4_F32 on a single wave.
// ---------------------------------------------------------------------------

#define N_RAYS 65536
#define N_SAMP 64
#define P_TOTAL (N_RAYS * N_SAMP)
#define HID 64

typedef __attribute__((ext_vector_type(2))) float v2f;
typedef __attribute__((ext_vector_type(8))) float v8f;

// ---------------------------------------------------------------------------
// c = w1 @ w2 via V_WMMA_F32_16X16X4_F32 (one wave32).
// A (16x4 f32): lane L (L<16: M=L, K in {0,1}; L>=16: M=L-16, K in {2,3}),
//   vgpr0 = K even, vgpr1 = K odd.  Set A[m,k] = w1[f, 4m+k].
// B (4x16 f32): symmetric layout with N in place of M. Set B[k,n] = w2[4n+k].
// Then D[m,n] = sum_k w1[f,4m+k] * w2[4n+k]  =>  trace(D) = c[f].
// C/D f32 16x16 layout: lane 0-15 -> M = vgpr, N = lane; lane 16-31 ->
//   M = vgpr + 8, N = lane - 16.  Diagonal: lanes 0..7 (v = L) and
//   lanes 24..31 (v = L - 24).
// ---------------------------------------------------------------------------
__global__ __launch_bounds__(32) void kp_compute_c_wmma(
    const float* __restrict__ w1,   // [2,64]
    const float* __restrict__ w2,   // [64,1]
    float* __restrict__ c_out)      // [2]
{
    const int lane  = (int)threadIdx.x;           // 0..31
    const int m     = lane & 15;
    const int khalf = lane >> 4;                  // 0 or 1
    const int kbase = 2 * khalf;

#if __has_builtin(__builtin_amdgcn_wmma_f32_16x16x4_f32)
    v2f b;
    b.x = w2[4 * m + kbase + 0];
    b.y = w2[4 * m + kbase + 1];

    float cacc[2];
#pragma unroll
    for (int f = 0; f < 2; ++f) {
        v2f a;
        a.x = w1[f * HID + 4 * m + kbase + 0];
        a.y = w1[f * HID + 4 * m + kbase + 1];
        v8f cz = {};
        v8f d = __builtin_amdgcn_wmma_f32_16x16x4_f32(
            /*neg_a=*/false, a, /*neg_b=*/false, b,
            /*c_mod=*/(short)0, cz, /*reuse_a=*/false, /*reuse_b=*/false);

        // extract this lane's diagonal element (if any)
        const int mbase = (lane >= 16) ? 8 : 0;
        const int n     = lane & 15;
        float diag = 0.0f;
#pragma unroll
        for (int v = 0; v < 8; ++v)
            if (v + mbase == n) diag = d[v];

        // wave32 all-reduce
#pragma unroll
        for (int off = 16; off >= 1; off >>= 1)
            diag += __shfl_xor(diag, off, 32);
        cacc[f] = diag;
    }
    if (lane == 0) { c_out[0] = cacc[0]; c_out[1] = cacc[1]; }
#else
    // Fallback: scalar wave reduction (should not be needed on gfx1250).
    float p0 = w1[lane] * w2[lane] + w1[lane + 32] * w2[lane + 32];
    float p1 = w1[HID + lane] * w2[lane] + w1[HID + lane + 32] * w2[lane + 32];
#pragma unroll
    for (int off = 16; off >= 1; off >>= 1) {
        p0 += __shfl_xor(p0, off, 32);
        p1 += __shfl_xor(p1, off, 32);
    }
    if (lane == 0) { c_out[0] = p0; c_out[1] = p1; }
#endif
}

// ---------------------------------------------------------------------------
// Repack one plane [2,H,W] (channel-split) -> [H*W] float2 (interleaved).
// ---------------------------------------------------------------------------
__global__ __launch_bounds__(256) void kp_repack(
    const float* __restrict__ src, float2* __restrict__ dst, int hw)
{
    int i = blockIdx.x * 256 + (int)threadIdx.x;
    if (i < hw) dst[i] = make_float2(src[i], src[i + hw]);
}

// ---------------------------------------------------------------------------
// Bilinear sample, interleaved float2 plane. Matches reference clamping:
//   x = clip((u+1)*0.5*(W-1), 0, W-1); x0 = clip(floor(x), 0, W-2); wx = x-x0
// ---------------------------------------------------------------------------
__device__ __forceinline__ float2 kp_bilerp2(
    const float2* __restrict__ pl, int W, int H, float u, float v)
{
    float x = (u + 1.0f) * 0.5f * (float)(W - 1);
    float y = (v + 1.0f) * 0.5f * (float)(H - 1);
    x = fminf(fmaxf(x, 0.0f), (float)(W - 1));
    y = fminf(fmaxf(y, 0.0f), (float)(H - 1));
    int x0 = min((int)x, W - 2);   // x >= 0, so (int) == floor
    int y0 = min((int)y, H - 2);
    float wx = x - (float)x0;
    float wy = y - (float)y0;

    const float2* r0 = pl + (size_t)y0 * (size_t)W + (size_t)x0;
    const float2* r1 = r0 + W;
    float2 v00 = r0[0], v01 = r0[1], v10 = r1[0], v11 = r1[1];

    float omx = 1.0f - wx, omy = 1.0f - wy;
    float tx = v00.x * omx + v01.x * wx;
    float ty = v00.y * omx + v01.y * wx;
    float bx = v10.x * omx + v11.x * wx;
    float by = v10.y * omx + v11.y * wx;
    return make_float2(tx * omy + bx * wy, ty * omy + by * wy);
}

// Same, but channel-split [2,H,W] layout (fallback path, 8 scalar loads).
__device__ __forceinline__ float2 kp_bilerp_split(
    const float* __restrict__ pl, int W, int H, float u, float v)
{
    float x = (u + 1.0f) * 0.5f * (float)(W - 1);
    float y = (v + 1.0f) * 0.5f * (float)(H - 1);
    x = fminf(fmaxf(x, 0.0f), (float)(W - 1));
    y = fminf(fmaxf(y, 0.0f), (float)(H - 1));
    int x0 = min((int)x, W - 2);
    int y0 = min((int)y, H - 2);
    float wx = x - (float)x0;
    float wy = y - (float)y0;
    float omx = 1.0f - wx, omy = 1.0f - wy;

    size_t hw = (size_t)W * (size_t)H;
    const float* b0 = pl + (size_t)y0 * (size_t)W + (size_t)x0;
    const float* b1 = b0 + hw;

    float t0 = b0[0] * omx + b0[1] * wx;
    float q0 = b0[W] * omx + b0[W + 1] * wx;
    float t1 = b1[0] * omx + b1[1] * wx;
    float q1 = b1[W] * omx + b1[W + 1] * wx;
    return make_float2(t0 * omy + q0 * wy, t1 * omy + q1 * wy);
}

// ---------------------------------------------------------------------------
// Main kernel: one thread per point. 1 x b128 point load, 24 x b64 gathers
// (L2-resident), dot with c, v_exp. Wave32 latency hiding via 4.19M threads.
// ---------------------------------------------------------------------------
__global__ __launch_bounds__(256) void kp_density_fast(
    const float* __restrict__ pts,
    const float2* __restrict__ pl0, const float2* __restrict__ pl1,
    const float2* __restrict__ pl2, const float2* __restrict__ pl3,
    const float2* __restrict__ pl4, const float2* __restrict__ pl5,
    const float* __restrict__ c_in,
    const float* __restrict__ aabb,
    float* __restrict__ out)
{
    int p = blockIdx.x * 256 + (int)threadIdx.x;
    if (p >= P_TOTAL) return;

    float4 pt = reinterpret_cast<const float4*>(pts)[p];
    float ptv[4] = { pt.x, pt.y, pt.z, pt.w };

    float pn[4];
#pragma unroll
    for (int c = 0; c < 4; ++c) {
        float a0 = aabb[c], a1 = aabb[4 + c];
        pn[c] = (ptv[c] - a0) * (2.0f / (a1 - a0)) - 1.0f;
    }

    float fx = 1.0f, fy = 1.0f;
    float2 s;
    // PAIRS = (0,1),(0,2),(0,3),(1,2),(1,3),(2,3); RES = [512,512,512,300]
    s = kp_bilerp2(pl0, 512, 512, pn[0], pn[1]); fx *= s.x; fy *= s.y;
    s = kp_bilerp2(pl1, 512, 512, pn[0], pn[2]); fx *= s.x; fy *= s.y;
    s = kp_bilerp2(pl2, 512, 300, pn[0], pn[3]); fx *= s.x; fy *= s.y;
    s = kp_bilerp2(pl3, 512, 512, pn[1], pn[2]); fx *= s.x; fy *= s.y;
    s = kp_bilerp2(pl4, 512, 300, pn[1], pn[3]); fx *= s.x; fy *= s.y;
    s = kp_bilerp2(pl5, 512, 300, pn[2], pn[3]); fx *= s.x; fy *= s.y;

    float c0 = c_in[0], c1 = c_in[1];
    out[p] = expf(fx * c0 + fy * c1);
}

// Fallback: no workspace repack; reads split layout, computes c inline
// (uniform loads -> scalar; 128 extra MACs per thread, still correct).
__global__ __launch_bounds__(256) void kp_density_fallback(
    const float* __restrict__ pts,
    const float* __restrict__ pl0, const float* __restrict__ pl1,
    const float* __restrict__ pl2, const float* __restrict__ pl3,
    const float* __restrict__ pl4, const float* __restrict__ pl5,
    const float* __restrict__ w1, const float* __restrict__ w2,
    const float* __restrict__ aabb,
    float* __restrict__ out)
{
    int p = blockIdx.x * 256 + (int)threadIdx.x;
    if (p >= P_TOTAL) return;

    float c0 = 0.0f, c1 = 0.0f;
#pragma unroll 8
    for (int h = 0; h < HID; ++h) {
        float w = w2[h];
        c0 += w1[h] * w;
        c1 += w1[HID + h] * w;
    }

    float4 pt = reinterpret_cast<const float4*>(pts)[p];
    float ptv[4] = { pt.x, pt.y, pt.z, pt.w };
    float pn[4];
#pragma unroll
    for (int c = 0; c < 4; ++c) {
        float a0 = aabb[c], a1 = aabb[4 + c];
        pn[c] = (ptv[c] - a0) * (2.0f / (a1 - a0)) - 1.0f;
    }

    float fx = 1.0f, fy = 1.0f;
    float2 s;
    s = kp_bilerp_split(pl0, 512, 512, pn[0], pn[1]); fx *= s.x; fy *= s.y;
    s = kp_bilerp_split(pl1, 512, 512, pn[0], pn[2]); fx *= s.x; fy *= s.y;
    s = kp_bilerp_split(pl2, 512, 300, pn[0], pn[3]); fx *= s.x; fy *= s.y;
    s = kp_bilerp_split(pl3, 512, 512, pn[1], pn[2]); fx *= s.x; fy *= s.y;
    s = kp_bilerp_split(pl4, 512, 300, pn[1], pn[3]); fx *= s.x; fy *= s.y;
    s = kp_bilerp_split(pl5, 512, 300, pn[2], pn[3]); fx *= s.x; fy *= s.y;

    out[p] = expf(fx * c0 + fy * c1);
}

// ---------------------------------------------------------------------------
extern "C" void kernel_launch(void* const* d_in, const int* in_sizes, int n_in,
                              void* d_out, int out_size, void* d_ws, size_t ws_size,
                              hipStream_t stream) {
    (void)in_sizes; (void)n_in; (void)out_size;

    const float* pts  = (const float*)d_in[0];
    const float* pl[6];
    for (int i = 0; i < 6; ++i) pl[i] = (const float*)d_in[1 + i];
    const float* w1   = (const float*)d_in[7];
    const float* w2   = (const float*)d_in[8];
    const float* aabb = (const float*)d_in[9];
    float* out = (float*)d_out;

    // plane texel counts: H*W for shapes (2, RES[k], RES[j])
    const size_t texels[6] = {
        512u * 512u, 512u * 512u, 300u * 512u,
        512u * 512u, 300u * 512u, 300u * 512u
    };

    // workspace layout: [0,16) -> c[2] (float2, padded); then repacked planes
    size_t off[6];
    size_t need = 16;
    for (int i = 0; i < 6; ++i) { off[i] = need; need += texels[i] * sizeof(float2); }

    const int pt_blocks = (P_TOTAL + 255) / 256;

    if (ws_size >= need) {
        float* c_ptr = (float*)d_ws;
        kp_compute_c_wmma<<<1, 32, 0, stream>>>(w1, w2, c_ptr);

        float2* rp[6];
        for (int i = 0; i < 6; ++i) {
            rp[i] = (float2*)((char*)d_ws + off[i]);
            int hw = (int)texels[i];
            kp_repack<<<(hw + 255) / 256, 256, 0, stream>>>(pl[i], rp[i], hw);
        }

        kp_density_fast<<<pt_blocks, 256, 0, stream>>>(
            pts, rp[0], rp[1], rp[2], rp[3], rp[4], rp[5],
            c_ptr, aabb, out);
    } else {
        kp_density_fallback<<<pt_blocks, 256, 0, stream>>>(
            pts, pl[0], pl[1], pl[2], pl[3], pl[4], pl[5],
            w1, w2, aabb, out);
    }
}